// PositionalEncoding_3341484556841
// MI455X (gfx1250) — compile-verified
//
#include <hip/hip_runtime.h>

#define N_NODES   8192
#define NUM_USERS 4096
#define LATENT    128
#define QDIM      32
#define COMB_DIM  224   // LATENT + 3*QDIM
#define DAMPING   0.85f
#define PTOL      1e-6f
#define MAX_ITER  100

#define NSLICE          32
#define ROWS_PER_SLICE  (N_NODES / NSLICE)          // 256
#define JBLK            (N_NODES / (256 * 4))       // 8 column-blocks (256 thr * float4)

typedef float v2f __attribute__((ext_vector_type(2)));
typedef float v4f __attribute__((ext_vector_type(4)));
typedef float v8f __attribute__((ext_vector_type(8)));

// ---------------- PageRank machinery ----------------

__global__ void k_init(float* pr, int* flag) {
    int i = blockIdx.x * 256 + threadIdx.x;
    if (i < N_NODES) pr[i] = 1.0f / (float)N_NODES;
    if (i == 0) *flag = 0;
}

// deg[row] = sum_j adj[row][j];  outdeg = max(deg, 1)   (b128 streaming)
__global__ void k_rowsum(const float* __restrict__ adj, float* deg, float* outdeg) {
    __shared__ float red[256];
    const int row = blockIdx.x;
    const int tid = threadIdx.x;
    const v4f* rp = (const v4f*)(adj + (size_t)row * N_NODES);
    float s = 0.f;
#pragma unroll
    for (int j = tid; j < N_NODES / 4; j += 256) {
        v4f a = rp[j];
        s += (a.x + a.y) + (a.z + a.w);
    }
    red[tid] = s; __syncthreads();
    for (int off = 128; off > 0; off >>= 1) {
        if (tid < off) red[tid] += red[tid + off];
        __syncthreads();
    }
    if (tid == 0) { deg[row] = red[0]; outdeg[row] = fmaxf(red[0], 1.0f); }
}

__global__ void k_weights(const float* pr, const float* outdeg, float* w, const int* flag) {
    if (*flag) return;
    int i = blockIdx.x * 256 + threadIdx.x;
    w[i] = pr[i] / outdeg[i];
}

// Split-K matvec: slice s covers rows [s*256, (s+1)*256); each thread owns 4 columns.
// partial[s][j] = sum_{i in slice} adj[i][j] * w[i]
__global__ void k_matvec_part(const float* __restrict__ adj, const float* __restrict__ w,
                              float* __restrict__ partial, const int* flag) {
    if (*flag) return;
    __shared__ float wl[ROWS_PER_SLICE];
    const int tid   = threadIdx.x;
    const int slice = blockIdx.y;
    const int j0    = blockIdx.x * 1024 + tid * 4;
    const int row0  = slice * ROWS_PER_SLICE;
    wl[tid] = w[row0 + tid];            // 256 threads == 256 rows/slice
    __syncthreads();
    v4f acc = {0.f, 0.f, 0.f, 0.f};
    const float* base = adj + (size_t)row0 * N_NODES + j0;
#pragma unroll 4
    for (int t = 0; t < ROWS_PER_SLICE; ++t) {
        v4f a = *(const v4f*)(base + (size_t)t * N_NODES);
        float wt = wl[t];
        acc.x = fmaf(a.x, wt, acc.x);
        acc.y = fmaf(a.y, wt, acc.y);
        acc.z = fmaf(a.z, wt, acc.z);
        acc.w = fmaf(a.w, wt, acc.w);
    }
    *(v4f*)(partial + (size_t)slice * N_NODES + j0) = acc;
}

// Deterministic fixed-order slice combine + damping + per-block L1-delta partials
__global__ void k_combine(const float* __restrict__ partial, const float* __restrict__ pr,
                          float* __restrict__ new_pr, float* pblk, const int* flag) {
    if (*flag) return;
    __shared__ float lds[256];
    const int tid = threadIdx.x;
    const int j = blockIdx.x * 256 + tid;
    float s = 0.f;
#pragma unroll
    for (int sl = 0; sl < NSLICE; ++sl)
        s += partial[(size_t)sl * N_NODES + j];
    float np = (1.0f - DAMPING) / (float)N_NODES + DAMPING * s;
    new_pr[j] = np;
    lds[tid] = fabsf(np - pr[j]);
    __syncthreads();
    for (int off = 128; off > 0; off >>= 1) {
        if (tid < off) lds[tid] += lds[tid + off];
        __syncthreads();
    }
    if (tid == 0) pblk[blockIdx.x] = lds[0];
}

// deterministic final reduction + convergence decision
__global__ void k_reduce(const float* pblk, int* flag) {
    if (*flag) return;
    __shared__ float lds[32];
    int tid = threadIdx.x;
    lds[tid] = pblk[tid];
    __syncthreads();
    for (int off = 16; off > 0; off >>= 1) {
        if (tid < off) lds[tid] += lds[tid + off];
        __syncthreads();
    }
    if (tid == 0 && lds[0] < PTOL) *flag = 1;   // converged: freeze pr (pre-update value)
}

__global__ void k_update(float* pr, const float* new_pr, const int* flag) {
    if (*flag) return;
    int i = blockIdx.x * 256 + threadIdx.x;
    pr[i] = new_pr[i];
}

// ---------------- combined-feature staging ----------------

__global__ void k_fill_x(const float* __restrict__ x, float* __restrict__ comb) {
    int idx = blockIdx.x * 256 + threadIdx.x;   // < 4096*128
    int u = idx >> 7, c = idx & 127;
    comb[(size_t)u * COMB_DIM + c] = x[idx];
}

// spe input is identically zero -> h = relu(b1); constant row broadcast
__global__ void k_fill_spe(const float* __restrict__ b1, const float* __restrict__ w2,
                           const float* __restrict__ b2, float* __restrict__ comb) {
    int idx = blockIdx.x * 256 + threadIdx.x;   // < 4096*32
    int u = idx >> 5, q2 = idx & 31;
    float s = b2[q2];
#pragma unroll
    for (int q = 0; q < QDIM; ++q)
        s = fmaf(fmaxf(b1[q], 0.f), w2[q * QDIM + q2], s);
    comb[(size_t)u * COMB_DIM + LATENT + q2] = s;
}

// generic 1->Q relu ->Q MLP over a scalar-per-user input (de: degrees, pre: pagerank)
__global__ void k_fill_mlp(const float* __restrict__ vin,
                           const float* __restrict__ w1, const float* __restrict__ b1,
                           const float* __restrict__ w2, const float* __restrict__ b2,
                           float* __restrict__ comb, int col_off) {
    int u = blockIdx.x * 256 + threadIdx.x;     // < 4096
    float v = vin[u];
    float h[QDIM];
#pragma unroll
    for (int q = 0; q < QDIM; ++q)
        h[q] = fmaxf(fmaf(v, w1[q], b1[q]), 0.f);
    for (int q2 = 0; q2 < QDIM; ++q2) {
        float s = b2[q2];
#pragma unroll
        for (int q = 0; q < QDIM; ++q)
            s = fmaf(h[q], w2[q * QDIM + q2], s);
        comb[(size_t)u * COMB_DIM + col_off + q2] = s;
    }
}

// ---------------- WMMA f32 GEMM: C[4096,128] = act(A[4096,K] @ B[K,128] + bias) --------
// One wave32 owns one 16x16 C tile; K stepped by 4 via V_WMMA_F32_16X16X4_F32.
// A 16x4 layout: lanes 0-15 hold (M=l, K={0,1}), lanes 16-31 hold (M=l, K={2,3}).
// B 4x16 layout: lanes 0-15 hold (K={0,1}, N=l), lanes 16-31 hold (K={2,3}, N=l).
// C/D: VGPR r -> M = half*8 + r, N = l.
__global__ void k_gemm_wmma(const float* __restrict__ A, int lda, int K,
                            const float* __restrict__ B, const float* __restrict__ bias,
                            float* __restrict__ C, int do_relu) {
    const int wave  = (blockIdx.x * blockDim.x + threadIdx.x) >> 5;
    const int lane  = threadIdx.x & 31;
    const int tileM = wave >> 3;          // 128/16 = 8 N-tiles
    const int tileN = wave & 7;
    const int half  = lane >> 4;
    const int l     = lane & 15;
    const int row   = tileM * 16 + l;
    const int n     = tileN * 16 + l;

    v8f acc = {};
    const float* ap = A + (size_t)row * lda + 2 * half;       // (row, k0 + 2*half)
    const float* bp = B + (size_t)(2 * half) * LATENT + n;    // (k0 + 2*half, n)
    for (int k0 = 0; k0 < K; k0 += 4) {
        v2f a = *(const v2f*)(ap + k0);                       // K, K+1 contiguous
        v2f b;
        b.x = bp[(size_t)k0 * LATENT];
        b.y = bp[(size_t)(k0 + 1) * LATENT];
        acc = __builtin_amdgcn_wmma_f32_16x16x4_f32(
            false, a, false, b, (short)0, acc, false, false);
    }
#pragma unroll
    for (int r = 0; r < 8; ++r) {
        int m = tileM * 16 + half * 8 + r;
        float v = acc[r] + bias[n];
        if (do_relu) v = fmaxf(v, 0.f);
        C[(size_t)m * LATENT + n] = v;
    }
}

// ---------------- launcher ----------------

extern "C" void kernel_launch(void* const* d_in, const int* in_sizes, int n_in,
                              void* d_out, int out_size, void* d_ws, size_t ws_size,
                              hipStream_t stream) {
    const float* x       = (const float*)d_in[0];
    const float* adj     = (const float*)d_in[1];
    // spe_w1 (d_in[2]) unused: its input is identically zero
    const float* spe_b1  = (const float*)d_in[3];
    const float* spe_w2  = (const float*)d_in[4];
    const float* spe_b2  = (const float*)d_in[5];
    const float* de_w1   = (const float*)d_in[6];
    const float* de_b1   = (const float*)d_in[7];
    const float* de_w2   = (const float*)d_in[8];
    const float* de_b2   = (const float*)d_in[9];
    const float* pre_w1  = (const float*)d_in[10];
    const float* pre_b1  = (const float*)d_in[11];
    const float* pre_w2  = (const float*)d_in[12];
    const float* pre_b2  = (const float*)d_in[13];
    const float* comb_w1 = (const float*)d_in[14];
    const float* comb_b1 = (const float*)d_in[15];
    const float* comb_w2 = (const float*)d_in[16];
    const float* comb_b2 = (const float*)d_in[17];
    // node_indices == arange(NUM_USERS): first 4096 rows, hardcoded

    float* ws      = (float*)d_ws;
    float* deg     = ws;                 //   8192
    float* outdeg  = ws + 8192;          //   8192
    float* pr      = ws + 16384;         //   8192
    float* w       = ws + 24576;         //   8192
    float* newpr   = ws + 32768;         //   8192
    float* partial = ws + 40960;         //  32*8192 = 262144
    float* pblk    = ws + 303104;        //   32
    int*   flag    = (int*)(ws + 303136);
    float* comb    = ws + 303168;        //  4096*224 = 917504 (16B aligned)
    float* H       = comb + (size_t)NUM_USERS * COMB_DIM;   // 4096*128
    float* out     = (float*)d_out;

    // init + degree pass (one 256MB stream over adj)
    k_init<<<32, 256, 0, stream>>>(pr, flag);
    k_rowsum<<<N_NODES, 256, 0, stream>>>(adj, deg, outdeg);

    // faithful 100-iteration power loop with deterministic early-stop flag
    for (int it = 0; it < MAX_ITER; ++it) {
        k_weights<<<32, 256, 0, stream>>>(pr, outdeg, w, flag);
        k_matvec_part<<<dim3(JBLK, NSLICE), 256, 0, stream>>>(adj, w, partial, flag);
        k_combine<<<32, 256, 0, stream>>>(partial, pr, newpr, pblk, flag);
        k_reduce<<<1, 32, 0, stream>>>(pblk, flag);
        k_update<<<32, 256, 0, stream>>>(pr, newpr, flag);
    }

    // stage combined = [x | spe | de | pre]  (4096 x 224)
    k_fill_x  <<<2048, 256, 0, stream>>>(x, comb);
    k_fill_spe<<<512, 256, 0, stream>>>(spe_b1, spe_w2, spe_b2, comb);
    k_fill_mlp<<<16, 256, 0, stream>>>(deg, de_w1, de_b1, de_w2, de_b2, comb, LATENT + QDIM);
    k_fill_mlp<<<16, 256, 0, stream>>>(pr,  pre_w1, pre_b1, pre_w2, pre_b2, comb, LATENT + 2 * QDIM);

    // combined MLP via f32 WMMA: 2048 tiles, 4 waves/block -> 512 blocks
    k_gemm_wmma<<<512, 128, 0, stream>>>(comb, COMB_DIM, COMB_DIM, comb_w1, comb_b1, H, 1);
    k_gemm_wmma<<<512, 128, 0, stream>>>(H, LATENT, LATENT, comb_w2, comb_b2, out, 0);
}